// RNNVanilla_19619410608757
// MI455X (gfx1250) — compile-verified
//
#include <hip/hip_runtime.h>
#include <cmath>

// Problem sizes (fixed by the reference).
#define TT 512   // timesteps
#define BB 64    // batch
#define II 512   // input dim
#define HH 512   // hidden dim
#define NWG 32   // persistent workgroups for the scan (one per WGP slice of H)

typedef __attribute__((ext_vector_type(2))) float v2f;
typedef __attribute__((ext_vector_type(8))) float v8f;

// Full-precision CDNA5 matrix op: D(16x16,f32) = A(16x4,f32) * B(4x16,f32) + C
__device__ __forceinline__ v8f wmma_f32_k4(v2f a, v2f b, v8f c) {
  return __builtin_amdgcn_wmma_f32_16x16x4_f32(
      /*neg_a=*/false, a, /*neg_b=*/false, b,
      /*c_mod=*/(short)0, c, /*reuse_a=*/false, /*reuse_b=*/false);
}

// ---------------------------------------------------------------------------
// Phase 1: input_term[t,b,h] = X[t,b,:] . Wxh[h,:] + bh[h]
// Grid: (TB/64, HH/32), 256 threads (8 waves). Wave w -> C tile (m = w&3, nl = w>>2).
// Wxh slice (32 rows x 512) staged in LDS once per block.
// ---------------------------------------------------------------------------
__global__ __launch_bounds__(256) void rnn_input_gemm(
    const float* __restrict__ X, const float* __restrict__ Wxh,
    const float* __restrict__ bh, float* __restrict__ out) {
  __shared__ __align__(16) float sW[32 * II];  // 64 KB of LDS (320 KB/WGP on CDNA5)

  const int bx = blockIdx.x;   // 64-row block of T*B
  const int by = blockIdx.y;   // 32-col block of H

  // Stage 32 rows of Wxh into LDS (row-major, contiguous K).
  for (int i = threadIdx.x; i < 32 * II / 4; i += 256) {
    const int r = i >> 7;        // 128 float4 per 512-float row
    const int c4 = i & 127;
    ((float4*)sW)[i] = ((const float4*)(Wxh + (size_t)(by * 32 + r) * II))[c4];
  }
  __syncthreads();

  const int wave = threadIdx.x >> 5;
  const int lane = threadIdx.x & 31;
  const int m  = wave & 3;        // M-tile within the 64-row block
  const int nl = wave >> 2;       // N-tile (0/1) within the 32-col block
  const int l15  = lane & 15;
  const int koff = (lane >> 4) * 2;   // lanes 16-31 hold K+2, K+3

  const float* Arow = X + (size_t)(bx * 64 + m * 16 + l15) * II;
  const float* Brow = sW + (nl * 16 + l15) * II;   // B[k][n] = Wxh[n][k] -> contiguous in k

  v8f acc = {};
#pragma unroll 4
  for (int k = 0; k < II; k += 4) {
    v2f a = *(const v2f*)(Arow + k + koff);
    v2f b = *(const v2f*)(Brow + k + koff);
    acc = wmma_f32_k4(a, b, acc);
  }

  const int colC = by * 32 + nl * 16 + l15;
  const float bias = bh[colC];
  float* orow = out + (size_t)(bx * 64 + m * 16 + (lane >> 4) * 8) * HH + colC;
#pragma unroll
  for (int v = 0; v < 8; v++) {
    orow[(size_t)v * HH] = acc[v] + bias;   // C layout: VGPR v -> row v (+8 for hi lanes)
  }
}

// ---------------------------------------------------------------------------
// Zero the grid-barrier counter (graph-safe per-launch init).
// ---------------------------------------------------------------------------
__global__ void rnn_zero_counter(unsigned* c) {
  if (threadIdx.x == 0) *c = 0u;
}

// ---------------------------------------------------------------------------
// Phase 2: persistent scan. 32 blocks x 128 threads (4 waves).
// Block j owns H columns [j*16, j*16+16); wave m owns batch rows [m*16, m*16+16).
// Whh slice (16 rows x 512 = 32 KB) lives in LDS for all 512 steps.
// h_{t-1} is read in-place from out[(t-1)]; steps separated by a global barrier.
// ---------------------------------------------------------------------------
__global__ __launch_bounds__(128) void rnn_scan(
    const float* __restrict__ Whh, float* __restrict__ out,
    float* __restrict__ hlast, unsigned* __restrict__ counter) {
  __shared__ __align__(16) float sW[16 * HH];  // 32 KB

  const int j = blockIdx.x;
  for (int i = threadIdx.x; i < 16 * HH / 4; i += 128) {
    const int r = i >> 7;
    const int c4 = i & 127;
    ((float4*)sW)[i] = ((const float4*)(Whh + (size_t)(j * 16 + r) * HH))[c4];
  }
  __syncthreads();

  const int m = threadIdx.x >> 5;      // wave id == M-tile (batch rows)
  const int lane = threadIdx.x & 31;
  const int l15  = lane & 15;
  const int koff = (lane >> 4) * 2;

  const float* Brow = sW + l15 * HH;   // B[k][n] = Whh[n][k] -> contiguous in k
  const int colC = j * 16 + l15;

  for (int t = 0; t < TT; t++) {
    v8f acc = {};
    if (t > 0) {
      const float* hp = out + (size_t)(t - 1) * BB * HH + (size_t)(m * 16 + l15) * HH;
#pragma unroll 4
      for (int k = 0; k < HH; k += 4) {
        v2f a = *(const v2f*)(hp + k + koff);
        v2f b = *(const v2f*)(Brow + k + koff);
        acc = wmma_f32_k4(a, b, acc);
      }
    }

    // h_t = tanh(input_term + h_{t-1} @ Whh^T), written in place over input_term.
    float* op = out + (size_t)t * BB * HH + (size_t)(m * 16 + (lane >> 4) * 8) * HH + colC;
    float* lp = hlast + (size_t)(m * 16 + (lane >> 4) * 8) * HH + colC;
#pragma unroll
    for (int v = 0; v < 8; v++) {
      const float val = tanhf(op[(size_t)v * HH] + acc[v]);
      op[(size_t)v * HH] = val;
      if (t == TT - 1) lp[(size_t)v * HH] = val;
    }

    // Device-wide step barrier (monotonic counter; release/acquire fences).
    __threadfence();
    __syncthreads();
    if (threadIdx.x == 0) {
      atomicAdd(counter, 1u);
      const unsigned target = (unsigned)(t + 1) * NWG;
      while (atomicAdd(counter, 0u) < target) {
        __builtin_amdgcn_s_sleep(1);
      }
    }
    __syncthreads();
    __threadfence();
  }
}

extern "C" void kernel_launch(void* const* d_in, const int* in_sizes, int n_in,
                              void* d_out, int out_size, void* d_ws, size_t ws_size,
                              hipStream_t stream) {
  (void)in_sizes; (void)n_in; (void)out_size; (void)ws_size;
  const float* X   = (const float*)d_in[0];   // [T, B, I]
  const float* Wxh = (const float*)d_in[1];   // [H, I]
  const float* Whh = (const float*)d_in[2];   // [H, H]
  const float* bh  = (const float*)d_in[3];   // [H]
  float* out   = (float*)d_out;                         // [T, B, H] then [B, H]
  float* hlast = out + (size_t)TT * BB * HH;
  unsigned* counter = (unsigned*)d_ws;

  rnn_zero_counter<<<1, 32, 0, stream>>>(counter);

  dim3 g1((TT * BB) / 64, HH / 32);
  rnn_input_gemm<<<g1, 256, 0, stream>>>(X, Wxh, bh, out);

  rnn_scan<<<NWG, 128, 0, stream>>>(Whh, out, hlast, counter);
}